// MemoryEfficientSplade_11493332484748
// MI455X (gfx1250) — compile-verified
//
#include <hip/hip_runtime.h>
#include <hip/hip_bf16.h>
#include <cstdint>

// Problem constants (from reference): B=8, S=512, V=30522, top_k read from d_in[2].
#define B_  8
#define S_  512
#define V_  30522
#define CH_ 4                 // S-chunks for extra parallelism in phase 1
#define SCHUNK_ (S_ / CH_)    // 128
#define V2_ (V_ / 2)          // 15261 float2 columns (V is even, only 8B-alignable)
#define D_  8                 // async pipeline: rows per group (2 groups in flight)

#define ASYNC_WAIT_8() asm volatile("s_wait_asynccnt 0x8" ::: "memory")
#define ASYNC_WAIT_0() asm volatile("s_wait_asynccnt 0x0" ::: "memory")

// ---------------------------------------------------------------------------
// Phase 1: masked max over one S-chunk, data staged through LDS with the
// gfx1250 async copy engine (GLOBAL_LOAD_ASYNC_TO_LDS_B64, ASYNCcnt).
// Thread = (b, chunk, 2 consecutive v). mask[b,s] is wave-uniform per
// iteration -> masked rows issue no copy at all (no memory traffic).
// Each wave owns a private 4KB LDS slice (2 groups x 8 rows x 32 lanes x 8B);
// ping-pong keeps up to 16 rows of copies in flight per wave with zero VGPR
// pressure and no block barriers.
// ---------------------------------------------------------------------------
__global__ __launch_bounds__(256) void splade_phase1_maxred(
    const float* __restrict__ logits,   // [B,S,V]
    const int*   __restrict__ mask,     // [B,S]
    float*       __restrict__ partial)  // [CH][B][V]
{
    __shared__ float2 stage[8][2][D_][32];   // [wave][group][slot][lane] = 32KB

    int tid = blockIdx.x * blockDim.x + threadIdx.x;
    const int T = B_ * CH_ * V2_;
    if (tid >= T) return;                    // no block barriers in this kernel

    const int i2 = tid % V2_;
    const int bc = tid / V2_;
    const int c  = bc % CH_;
    const int b  = bc / CH_;
    const int v  = i2 * 2;
    const int s0 = c * SCHUNK_;

    const int wv = threadIdx.x >> 5;
    const int ln = threadIdx.x & 31;

    const int*    mrow = mask + b * S_;
    const float2* base = (const float2*)(logits + (size_t)(b * S_ + s0) * V_ + v);

    float m0 = -INFINITY, m1 = -INFINITY;

    int g = 0, cur = 0, prevG = 0;
    bool havePrev = false;

    for (int s = 0; s < SCHUNK_; ++s) {
        if (mrow[s0 + s] != 0) {
            // generic shared pointer: low 32 bits = block-relative LDS byte addr
            const unsigned lds = (unsigned)(unsigned long long)(&stage[wv][g][cur][ln]);
            const float2* gp = base + (size_t)s * V2_;
            asm volatile("global_load_async_to_lds_b64 %0, %1, off"
                         :: "v"(lds), "v"(gp) : "memory");
            if (++cur == D_) {
                if (havePrev) {
                    ASYNC_WAIT_8();          // in-order: older group has landed
#pragma unroll
                    for (int j = 0; j < D_; ++j) {
                        const float2 x = stage[wv][prevG][j][ln];
                        m0 = fmaxf(m0, x.x);
                        m1 = fmaxf(m1, x.y);
                    }
                }
                havePrev = true;
                prevG = g;
                g ^= 1;
                cur = 0;
            }
        }
    }

    // Drain: everything outstanding, then consume the last full + partial group.
    ASYNC_WAIT_0();
    if (havePrev) {
#pragma unroll
        for (int j = 0; j < D_; ++j) {
            const float2 x = stage[wv][prevG][j][ln];
            m0 = fmaxf(m0, x.x);
            m1 = fmaxf(m1, x.y);
        }
    }
    for (int j = 0; j < cur; ++j) {
        const float2 x = stage[wv][g][j][ln];
        m0 = fmaxf(m0, x.x);
        m1 = fmaxf(m1, x.y);
    }

    float* p = partial + ((size_t)c * B_ + b) * V_ + v;
    p[0] = m0;
    p[1] = m1;
}

// ---------------------------------------------------------------------------
// Phase 2: combine CH partials, apply log1p(relu(.)), write values to d_out.
// ---------------------------------------------------------------------------
__global__ __launch_bounds__(256) void splade_phase2_activate(
    const float* __restrict__ partial,  // [CH][B][V]
    float*       __restrict__ values)   // [B,V] (= d_out)
{
    int tid = blockIdx.x * blockDim.x + threadIdx.x;
    const int T = B_ * V2_;
    if (tid >= T) return;

    const int i2 = tid % V2_;
    const int b  = tid / V2_;
    const int v  = i2 * 2;

    float m0 = -INFINITY, m1 = -INFINITY;
#pragma unroll
    for (int c = 0; c < CH_; ++c) {
        const float2 x = *(const float2*)(partial + ((size_t)c * B_ + b) * V_ + v);
        m0 = fmaxf(m0, x.x);
        m1 = fmaxf(m1, x.y);
    }
    float2 r;
    r.x = log1pf(fmaxf(m0, 0.0f));   // relu folded into fmax; -inf -> 0
    r.y = log1pf(fmaxf(m1, 0.0f));
    *(float2*)(values + (size_t)b * V_ + v) = r;
}

// ---------------------------------------------------------------------------
// Phase 3: per batch row, exact k-th largest via 31-step bit search on the
// (non-negative) float bit patterns, then threshold the row in place.
// One 1024-thread block per row; ~30 values cached in registers per thread.
// ---------------------------------------------------------------------------
__global__ __launch_bounds__(1024) void splade_phase3_topk(
    float*     __restrict__ values,    // [B,V] in/out (= d_out)
    const int* __restrict__ topk_ptr)  // scalar k
{
    const int b = blockIdx.x;
    const int t = threadIdx.x;
    const int k = topk_ptr[0];

    float* row = values + (size_t)b * V_;

    constexpr int J = (V_ + 1023) / 1024;  // 30
    float vals[J];
#pragma unroll
    for (int j = 0; j < J; ++j) {
        const int idx = t + j * 1024;
        vals[j] = (idx < V_) ? row[idx] : 0.0f;  // pad bits = 0, never counted
    }

    __shared__ int s_cnt;
    unsigned ans = 0u;

    for (int bit = 30; bit >= 0; --bit) {        // bit 31 is sign: values >= 0
        const unsigned trial = ans | (1u << bit);
        int c = 0;
#pragma unroll
        for (int j = 0; j < J; ++j)
            c += (__float_as_uint(vals[j]) >= trial) ? 1 : 0;

        // wave32 reduction, then one LDS atomic per wave
#pragma unroll
        for (int off = 16; off > 0; off >>= 1)
            c += __shfl_down(c, off, 32);

        __syncthreads();                 // protect previous iteration's read
        if (t == 0) s_cnt = 0;
        __syncthreads();
        if ((t & 31) == 0) atomicAdd(&s_cnt, c);
        __syncthreads();
        if (s_cnt >= k) ans = trial;     // keep largest u with count(>=u) >= k
    }

    const float thr = __uint_as_float(ans);  // exact k-th largest value
#pragma unroll
    for (int j = 0; j < J; ++j) {
        const int idx = t + j * 1024;
        if (idx < V_) row[idx] = (vals[j] >= thr) ? vals[j] : 0.0f;
    }
}

// ---------------------------------------------------------------------------
extern "C" void kernel_launch(void* const* d_in, const int* in_sizes, int n_in,
                              void* d_out, int out_size, void* d_ws, size_t ws_size,
                              hipStream_t stream)
{
    const float* logits = (const float*)d_in[0];   // [8,512,30522] f32
    const int*   mask   = (const int*)d_in[1];     // [8,512] i32
    const int*   topk   = (const int*)d_in[2];     // scalar
    float*       out    = (float*)d_out;           // [8,30522] f32
    float*       part   = (float*)d_ws;            // CH*B*V floats ≈ 3.9 MB

    const int T1 = B_ * CH_ * V2_;                 // 488,352 threads
    splade_phase1_maxred<<<(T1 + 255) / 256, 256, 0, stream>>>(logits, mask, part);

    const int T2 = B_ * V2_;                       // 122,088 threads
    splade_phase2_activate<<<(T2 + 255) / 256, 256, 0, stream>>>(part, out);

    splade_phase3_topk<<<B_, 1024, 0, stream>>>(out, topk);
}